// MessagePassingLayer_78305843741159
// MI455X (gfx1250) — compile-verified
//
#include <hip/hip_runtime.h>
#include <hip/hip_bf16.h>

#define EMBED   256
#define KDIM    512
#define M_TILES 4
#define TILE_M  (16 * M_TILES)   // 64 edges per block
#define LDS_ROW 516              // 512 + pad(4): lane m starts at bank (4*m)%64 -> conflict-free

typedef float v2f __attribute__((ext_vector_type(2)));
typedef float v8f __attribute__((ext_vector_type(8)));

// ---------------------------------------------------------------------------
// Kernel 1: per-node message counts (degree in both directions)
// ---------------------------------------------------------------------------
__global__ __launch_bounds__(256) void count_kernel(const long long* __restrict__ ht,
                                                    float* __restrict__ counts,
                                                    int nEdges) {
    int e = blockIdx.x * blockDim.x + threadIdx.x;
    if (e < nEdges) {
        int h  = (int)ht[(size_t)e * 2 + 0];
        int tl = (int)ht[(size_t)e * 2 + 1];
        atomicAdd(&counts[tl], 1.0f);   // msg_fwd lands on tail
        atomicAdd(&counts[h],  1.0f);   // msg_back lands on head
    }
}

// ---------------------------------------------------------------------------
// Kernel 2: fused gather + GEMM (fp32 WMMA) + bias + scatter-add
//   grid.x = edge blocks of 64, grid.y = direction (0 = fwd, 1 = back)
//   block  = 256 threads = 8 waves; wave w owns output columns [32w, 32w+32)
//   Each wave: 4 M-tiles x 2 N-tiles = 8 fp32 C-tiles (64 acc VGPRs).
//   B fragments (weights) are reused across 4 M-tiles -> 4x less L2 traffic.
// ---------------------------------------------------------------------------
__global__ __launch_bounds__(256) void gemm_scatter(const float* __restrict__ H,
                                                    const float* __restrict__ E,
                                                    const long long* __restrict__ ht,
                                                    const float* __restrict__ Wf,
                                                    const float* __restrict__ bf,
                                                    const float* __restrict__ Wb,
                                                    const float* __restrict__ bb,
                                                    float* __restrict__ agg,
                                                    int nEdges) {
    __shared__ float feat[TILE_M * LDS_ROW];  // 64 x 516 fp32 ~ 129 KB
    __shared__ int   srcIdx[TILE_M];
    __shared__ int   dstIdx[TILE_M];

    const int dir   = blockIdx.y;
    const int eBase = blockIdx.x * TILE_M;
    const int t     = threadIdx.x;

    if (t < TILE_M) {
        int e  = eBase + t;
        int ec = e < nEdges ? e : (nEdges - 1);
        int h  = (int)ht[(size_t)ec * 2 + 0];
        int tl = (int)ht[(size_t)ec * 2 + 1];
        srcIdx[t] = (dir == 0) ? h : tl;
        dstIdx[t] = (e < nEdges) ? ((dir == 0) ? tl : h) : -1;  // -1: skip scatter
    }
    __syncthreads();

    // Stage A-tile: 64 rows x 512 floats of [H[src] | E[e]] -> LDS (float4 chunks)
    #pragma unroll
    for (int c = 0; c < TILE_M * 128 / 256; ++c) {   // 32 chunks/thread
        int lin = c * 256 + t;              // float4 units
        int r   = lin >> 7;                 // row 0..63
        int c4  = lin & 127;                // float4 within row
        int e   = eBase + r;
        int ec  = e < nEdges ? e : (nEdges - 1);
        const float* src = (c4 < 64)
            ? (H + (size_t)srcIdx[r] * EMBED + c4 * 4)
            : (E + (size_t)ec * EMBED + (c4 - 64) * 4);
        float4 v = *(const float4*)src;
        *(float4*)&feat[r * LDS_ROW + c4 * 4] = v;
    }
    __syncthreads();

    const int wave  = t >> 5;
    const int lane  = t & 31;
    const int laneM = lane & 15;
    const int kOff  = (lane < 16) ? 0 : 2;   // ISA 16x4 fp32 A/B layout: hi lanes hold K=2,3

    const float* Wm   = (dir == 0) ? Wf : Wb;
    const float* bias = (dir == 0) ? bf : bb;

    const int n0 = wave * 32 + laneM;        // column of N-tile 0
    const int n1 = n0 + 16;                  // column of N-tile 1

    const float* w0 = Wm + (size_t)n0 * KDIM + kOff;   // B[k][n] = W[n][k]
    const float* w1 = Wm + (size_t)n1 * KDIM + kOff;

    const float* aBase[M_TILES];
    #pragma unroll
    for (int mt = 0; mt < M_TILES; ++mt)
        aBase[mt] = &feat[(mt * 16 + laneM) * LDS_ROW + kOff];

    v8f acc[M_TILES][2];
    #pragma unroll
    for (int mt = 0; mt < M_TILES; ++mt) {
        acc[mt][0] = (v8f){};
        acc[mt][1] = (v8f){};
    }

    #pragma unroll 4
    for (int kb = 0; kb < KDIM / 4; ++kb) {
        v2f b0 = *(const v2f*)(w0 + kb * 4);           // L2-resident weights
        v2f b1 = *(const v2f*)(w1 + kb * 4);
        #pragma unroll
        for (int mt = 0; mt < M_TILES; ++mt) {
            v2f a = *(const v2f*)(aBase[mt] + kb * 4); // ds_load_b64, conflict-free
            acc[mt][0] = __builtin_amdgcn_wmma_f32_16x16x4_f32(false, a, false, b0,
                                                               (short)0, acc[mt][0], false, false);
            acc[mt][1] = __builtin_amdgcn_wmma_f32_16x16x4_f32(false, a, false, b1,
                                                               (short)0, acc[mt][1], false, false);
        }
    }

    const float bv0 = bias[n0];
    const float bv1 = bias[n1];
    const int mBase = (lane < 16) ? 0 : 8;   // C layout: VGPR j = row j (+8 for hi lanes)

    #pragma unroll
    for (int mt = 0; mt < M_TILES; ++mt) {
        #pragma unroll
        for (int j = 0; j < 8; ++j) {
            int row = dstIdx[mt * 16 + mBase + j];
            if (row >= 0) {
                atomicAdd(&agg[(size_t)row * EMBED + n0], acc[mt][0][j] + bv0);
                atomicAdd(&agg[(size_t)row * EMBED + n1], acc[mt][1][j] + bv1);
            }
        }
    }
}

// ---------------------------------------------------------------------------
// Kernel 3: mean, leaky-relu, residual, LayerNorm.  One block per node.
// ---------------------------------------------------------------------------
__global__ __launch_bounds__(256) void finalize(const float* __restrict__ agg,
                                                const float* __restrict__ counts,
                                                const float* __restrict__ H,
                                                const float* __restrict__ lnw,
                                                const float* __restrict__ lnb,
                                                float* __restrict__ out) {
    const int n = blockIdx.x;
    const int t = threadIdx.x;

    float c = counts[n];
    float v = agg[(size_t)n * EMBED + t] / c;
    v = (v >= 0.0f) ? v : 0.01f * v;                  // leaky relu
    float x = v + H[(size_t)n * EMBED + t];           // residual

    __shared__ float red[256];
    red[t] = x;
    __syncthreads();
    #pragma unroll
    for (int s = 128; s > 0; s >>= 1) {
        if (t < s) red[t] += red[t + s];
        __syncthreads();
    }
    float mu = red[0] * (1.0f / EMBED);
    __syncthreads();

    float d = x - mu;
    red[t] = d * d;
    __syncthreads();
    #pragma unroll
    for (int s = 128; s > 0; s >>= 1) {
        if (t < s) red[t] += red[t + s];
        __syncthreads();
    }
    float var = red[0] * (1.0f / EMBED);
    float r   = rsqrtf(var + 1e-5f);

    out[(size_t)n * EMBED + t] = d * r * lnw[t] + lnb[t];
}

// ---------------------------------------------------------------------------
// Launch
// Inputs: 0:H 1:E 2:ht 3:queries 4:influence_weights 5:W_fwd 6:b_fwd
//         7:W_back 8:b_back 9:ln_w 10:ln_b
// ---------------------------------------------------------------------------
extern "C" void kernel_launch(void* const* d_in, const int* in_sizes, int n_in,
                              void* d_out, int out_size, void* d_ws, size_t ws_size,
                              hipStream_t stream) {
    const float*     H  = (const float*)d_in[0];
    const float*     E  = (const float*)d_in[1];
    const long long* ht = (const long long*)d_in[2];
    const float*     Wf = (const float*)d_in[5];
    const float*     bf = (const float*)d_in[6];
    const float*     Wb = (const float*)d_in[7];
    const float*     bb = (const float*)d_in[8];
    const float*     lw = (const float*)d_in[9];
    const float*     lb = (const float*)d_in[10];
    float*           out = (float*)d_out;

    const int nNodes = in_sizes[0] / EMBED;
    const int nEdges = in_sizes[2] / 2;

    float* agg    = (float*)d_ws;
    float* counts = agg + (size_t)nNodes * EMBED;
    size_t zeroBytes = (size_t)nNodes * EMBED * sizeof(float) + (size_t)nNodes * sizeof(float);
    (void)hipMemsetAsync(d_ws, 0, zeroBytes, stream);

    count_kernel<<<(nEdges + 255) / 256, 256, 0, stream>>>(ht, counts, nEdges);

    dim3 grid((nEdges + TILE_M - 1) / TILE_M, 2);
    gemm_scatter<<<grid, 256, 0, stream>>>(H, E, ht, Wf, bf, Wb, bb, agg, nEdges);

    finalize<<<nNodes, 256, 0, stream>>>(agg, counts, H, lw, lb, out);
}